// SolveScheduling_69080253989325
// MI455X (gfx1250) — compile-verified
//
#include <hip/hip_runtime.h>
#include <math.h>

// ---------------- problem constants (match reference) ----------------
#define NN        24      // horizon n
#define NM1       23      // n-1
#define MM        46      // 2*(n-1) constraints
#define SQP_TOTAL 21      // 20 SQP iterations + identical final solve
#define IPM_ITERS 18
#define TPB       32      // ONE wave32 per workgroup -> full VGPR budget per wave

typedef __attribute__((ext_vector_type(4))) unsigned int u32x4;
typedef __attribute__((ext_vector_type(4))) int          i32x4;
typedef __attribute__((ext_vector_type(8))) int          i32x8;

// ---------------- Tensor Data Mover (CDNA5 TDM) helpers ----------------
// 1-D tile: tensor == tile == `elems` fp32 (elems <= 65535, multiple of 4B).
__device__ __forceinline__ void tdm_desc(u32x4 &g0, i32x8 &g1,
                                         unsigned lds_byte_off,
                                         unsigned long long gaddr,
                                         unsigned elems) {
  // D# group 0 (128b): [1:0]=count=1, [63:32]=lds_addr,
  // [120:64]=global_addr, [127:126]=type=2 ("image")
  g0.x = 1u;
  g0.y = lds_byte_off;
  g0.z = (unsigned)(gaddr & 0xffffffffull);
  g0.w = (unsigned)((gaddr >> 32) & 0x01ffffffull) | (2u << 30);
  // D# group 1 (256b): wg_mask=0, data_size=2 (4 bytes), no pad/iterate
  g1[0] = (int)(2u << 16);                                 // data_size @ [17:16]
  g1[1] = (int)((elems & 0xffffu) << 16);                  // tensor_dim0[15:0] @ [63:48]
  g1[2] = (int)(((elems >> 16) & 0xffffu) | (1u << 16));   // tensor_dim0 hi, tensor_dim1=1
  g1[3] = (int)((elems & 0xffffu) << 16);                  // tile_dim0 @ [127:112]
  g1[4] = 1;                                               // tile_dim1=1, tile_dim2=0
  g1[5] = (int)elems;                                      // tensor_dim0_stride (1 row, unused)
  g1[6] = 0;
  g1[7] = 0;
}

__device__ __forceinline__ void tdm_load(u32x4 g0, i32x8 g1) {
  i32x4 z4 = {0, 0, 0, 0};
#if defined(__clang_major__) && (__clang_major__ >= 23)
  i32x8 z8 = {0, 0, 0, 0, 0, 0, 0, 0};
  __builtin_amdgcn_tensor_load_to_lds(g0, g1, z4, z4, z8, 0);
#else
  __builtin_amdgcn_tensor_load_to_lds(g0, g1, z4, z4, 0);
#endif
}

__device__ __forceinline__ void tdm_store(u32x4 g0, i32x8 g1) {
  i32x4 z4 = {0, 0, 0, 0};
#if defined(__clang_major__) && (__clang_major__ >= 23)
  i32x8 z8 = {0, 0, 0, 0, 0, 0, 0, 0};
  __builtin_amdgcn_tensor_store_from_lds(g0, g1, z4, z4, z8, 0);
#else
  __builtin_amdgcn_tensor_store_from_lds(g0, g1, z4, z4, 0);
#endif
}

// ---------------- small per-lane linear algebra ----------------
// LDL^T solve of the tridiagonal Schur matrix (unit-bidiagonal L, diag D)
__device__ __forceinline__ void ldl_solve(float *x, const float *lfac, const float *dinv) {
#pragma unroll
  for (int j = 1; j < NN; ++j) x[j] -= lfac[j - 1] * x[j - 1];
#pragma unroll
  for (int j = 0; j < NN; ++j) x[j] *= dinv[j];
#pragma unroll
  for (int j = NN - 2; j >= 0; --j) x[j] -= lfac[j] * x[j + 1];
}

// rhs_j = -rd_j - (t@G)_j where (t@G)_j = u_j - u_{j-1}, u_j = t_j - t_{j+23}
__device__ __forceinline__ void build_rhs(float *dz, const float *u, const float *rd) {
#pragma unroll
  for (int j = 0; j < NN; ++j) {
    float gu = 0.0f;
    if (j < NM1) gu += u[j];
    if (j > 0)   gu -= u[j - 1];
    dz[j] = -rd[j] - gu;
  }
}

__device__ __forceinline__ float frac_alpha(float a, float x, float dx) {
  return (dx < 0.0f) ? fminf(a, -x / dx) : a;
}

// ---------------- main kernel: one batch element per lane ----------------
__global__ __launch_bounds__(TPB, 1)
void solve_scheduling_kernel(const float *__restrict__ mu_g,
                             const float *__restrict__ sig_g,
                             float *__restrict__ out_g) {
  __shared__ float sm_mu[TPB * NN];   // 3 KB
  __shared__ float sm_sig[TPB * NN];  // 3 KB
  const int tid = threadIdx.x;
  const unsigned elems = (unsigned)(TPB * NN);  // 768 fp32 per wave tile
  const unsigned long long byte_off =
      (unsigned long long)blockIdx.x * (unsigned long long)elems * 4ull;

  // ---- TDM: DMA mu/sig tiles global -> LDS (this wave issues its own) ----
  {
    u32x4 g0; i32x8 g1;
    tdm_desc(g0, g1, (unsigned)(unsigned long long)(void *)sm_mu,
             (unsigned long long)(const void *)mu_g + byte_off, elems);
    tdm_load(g0, g1);
    tdm_desc(g0, g1, (unsigned)(unsigned long long)(void *)sm_sig,
             (unsigned long long)(const void *)sig_g + byte_off, elems);
    tdm_load(g0, g1);
    __builtin_amdgcn_s_wait_tensorcnt(0);
  }
  __syncthreads();

  // ---- register-resident SQP + Mehrotra PDIPM ----
  float z[NN], q[NN], p[NN];
#pragma unroll
  for (int j = 0; j < NN; ++j) z[j] = sm_mu[tid * NN + j];  // z0 = mu

#pragma unroll 1
  for (int it = 0; it < SQP_TOTAL; ++it) {
    // GLinear/GQuadratic approx around current z  ->  q = g''+1, p = g'-g''z-mu
#pragma unroll
    for (int j = 0; j < NN; ++j) {
      float m  = sm_mu[tid * NN + j];
      float sd = sm_sig[tid * NN + j];
      float sdi = 1.0f / sd;
      float t  = (z[j] - m) * sdi;
      float pdf = 0.3989422804014327f * expf(-0.5f * t * t) * sdi;
      float cdf = 0.5f * erfcf(-t * 0.7071067811865476f);
      float dg  = 50.5f * cdf - 50.0f;   // (gu+go)*Phi - gu
      float d2g = 50.5f * pdf;           // (gu+go)*N
      q[j] = d2g + 1.0f;
      p[j] = dg - d2g * z[j] - m;
    }

    // QP init: z = -p/q ; s = max(h - Gz, 1) ; lam = 1
    float s[MM], lam[MM];
#pragma unroll
    for (int j = 0; j < NN; ++j) z[j] = -p[j] / q[j];
#pragma unroll
    for (int i = 0; i < NM1; ++i) {
      float d = z[i] - z[i + 1];
      s[i]       = fmaxf(0.4f - d, 1.0f);
      s[i + NM1] = fmaxf(0.4f + d, 1.0f);
      lam[i] = 1.0f; lam[i + NM1] = 1.0f;
    }

#pragma unroll 1
    for (int k = 0; k < IPM_ITERS; ++k) {
      float sinv[MM];
#pragma unroll
      for (int i = 0; i < MM; ++i) sinv[i] = 1.0f / s[i];

      // residuals + duality gap (G is a +/- first-difference stencil)
      float rp[MM];
      float gap = 0.0f;
#pragma unroll
      for (int i = 0; i < NM1; ++i) {
        float d = z[i] - z[i + 1];
        rp[i]       =  d + s[i]       - 0.4f;
        rp[i + NM1] = -d + s[i + NM1] - 0.4f;
        gap += s[i] * lam[i] + s[i + NM1] * lam[i + NM1];
      }
      gap *= (1.0f / MM);

      float rd[NN];
#pragma unroll
      for (int j = 0; j < NN; ++j) {
        float lg = 0.0f;
        if (j < NM1) lg += lam[j] - lam[j + NM1];
        if (j > 0)   lg -= lam[j - 1] - lam[j - 1 + NM1];
        rd[j] = q[j] * z[j] + p[j] + lg;
      }

      // Schur matrix M = diag(q) + D' diag(v) D is TRIDIAGONAL: LDL^T factor
      float v[NN];
#pragma unroll
      for (int i = 0; i < NM1; ++i)
        v[i] = lam[i] * sinv[i] + lam[i + NM1] * sinv[i + NM1];
      float lfac[NM1], dinv[NN];
      dinv[0] = 1.0f / (q[0] + v[0]);
#pragma unroll
      for (int j = 0; j < NM1; ++j) {
        float lj = -v[j] * dinv[j];            // l_j = e_j/d_j, e_j = -v_j
        float diag = q[j + 1] + v[j];
        if (j + 1 < NM1) diag += v[j + 1];
        float dj1 = diag + lj * v[j];          // d_{j+1} = diag - v_j^2/d_j
        dinv[j + 1] = 1.0f / dj1;
        lfac[j] = lj;
      }

      // ---- predictor: rc = lam*s ----
      float dza[NN];
      {
        float u[NN];
#pragma unroll
        for (int i = 0; i < NM1; ++i) {
          float t1 = lam[i]       * (rp[i]       - s[i])       * sinv[i];
          float t2 = lam[i + NM1] * (rp[i + NM1] - s[i + NM1]) * sinv[i + NM1];
          u[i] = t1 - t2;
        }
        build_rhs(dza, u, rd);
        ldl_solve(dza, lfac, dinv);
      }
      // alpha_aff + moment accumulation; keep only prod = ds_a*dl_a for the
      // corrector. gap_aff = gap + (a*S1 + a^2*S2)/46 in closed form.
      float prod[MM];
      float aaff = 1.0f, S1 = 0.0f, S2 = 0.0f;
#pragma unroll
      for (int i = 0; i < NM1; ++i) {
        float gd = dza[i] - dza[i + 1];        // (G dz)_i ; row i+23 is -gd
        float d1 = -rp[i] - gd;
        float l1 = lam[i] * (rp[i] - s[i] + gd) * sinv[i];
        float d2 = -rp[i + NM1] + gd;
        float l2 = lam[i + NM1] * (rp[i + NM1] - s[i + NM1] - gd) * sinv[i + NM1];
        float pr1 = d1 * l1, pr2 = d2 * l2;
        prod[i] = pr1; prod[i + NM1] = pr2;
        S1 += s[i] * l1 + lam[i] * d1 + s[i + NM1] * l2 + lam[i + NM1] * d2;
        S2 += pr1 + pr2;
        aaff = frac_alpha(aaff, s[i], d1);
        aaff = frac_alpha(aaff, lam[i], l1);
        aaff = frac_alpha(aaff, s[i + NM1], d2);
        aaff = frac_alpha(aaff, lam[i + NM1], l2);
      }
      float gaff = gap + (aaff * S1 + aaff * aaff * S2) * (1.0f / MM);
      float rr  = gaff / gap;
      float smg = rr * rr * rr * gap;          // sigma * gap

      // ---- corrector: rc = lam*s + prod - sigma*gap ----
      float dzc[NN];
      {
        float u[NN];
#pragma unroll
        for (int i = 0; i < NM1; ++i) {
          float t1 = (lam[i] * (rp[i] - s[i]) - prod[i] + smg) * sinv[i];
          float t2 = (lam[i + NM1] * (rp[i + NM1] - s[i + NM1])
                      - prod[i + NM1] + smg) * sinv[i + NM1];
          u[i] = t1 - t2;
        }
        build_rhs(dzc, u, rd);
        ldl_solve(dzc, lfac, dinv);
      }
      // corrector steps + final step length (store: registers are plentiful
      // now that one wave owns the whole SIMD register file)
      float dsc[MM], dlc[MM];
      float a = 1.0f;
#pragma unroll
      for (int i = 0; i < NM1; ++i) {
        float gd = dzc[i] - dzc[i + 1];
        float d1 = -rp[i] - gd;
        float l1 = (lam[i] * (rp[i] - s[i] + gd) - prod[i] + smg) * sinv[i];
        float d2 = -rp[i + NM1] + gd;
        float l2 = (lam[i + NM1] * (rp[i + NM1] - s[i + NM1] - gd)
                    - prod[i + NM1] + smg) * sinv[i + NM1];
        dsc[i] = d1; dlc[i] = l1; dsc[i + NM1] = d2; dlc[i + NM1] = l2;
        a = frac_alpha(a, s[i], d1);
        a = frac_alpha(a, lam[i], l1);
        a = frac_alpha(a, s[i + NM1], d2);
        a = frac_alpha(a, lam[i + NM1], l2);
      }
      a *= 0.99f;

#pragma unroll
      for (int i = 0; i < MM; ++i) { s[i] += a * dsc[i]; lam[i] += a * dlc[i]; }
#pragma unroll
      for (int j = 0; j < NN; ++j) z[j] += a * dzc[j];
    } // IPM
  }   // SQP (+ final solve)

  // ---- stage result into LDS, TDM DMA LDS -> global ----
#pragma unroll
  for (int j = 0; j < NN; ++j) sm_mu[tid * NN + j] = z[j];
  __syncthreads();   // LDS fence: lane stores visible before TDM reads LDS
  {
    u32x4 g0; i32x8 g1;
    tdm_desc(g0, g1, (unsigned)(unsigned long long)(void *)sm_mu,
             (unsigned long long)(void *)out_g + byte_off, elems);
    tdm_store(g0, g1);
    __builtin_amdgcn_s_wait_tensorcnt(0);
  }
}

extern "C" void kernel_launch(void* const* d_in, const int* in_sizes, int n_in,
                              void* d_out, int out_size, void* d_ws, size_t ws_size,
                              hipStream_t stream) {
  (void)n_in; (void)out_size; (void)d_ws; (void)ws_size;
  const float* mu  = (const float*)d_in[0];
  const float* sig = (const float*)d_in[1];
  float* out = (float*)d_out;
  const int rows   = in_sizes[0] / NN;   // 8192 batch elements
  const int blocks = rows / TPB;         // 256 single-wave workgroups
  hipLaunchKernelGGL(solve_scheduling_kernel, dim3(blocks), dim3(TPB), 0, stream,
                     mu, sig, out);
}